// GINLayer_80633716015135
// MI455X (gfx1250) — compile-verified
//
#include <hip/hip_runtime.h>

typedef float v2f __attribute__((ext_vector_type(2)));
typedef float v8f __attribute__((ext_vector_type(8)));

#define N_NODES 50000
#define N_EDGES 800000
#define D 128
#define M_TILES (N_NODES / 16)   // 3125, exact

// ---------------------------------------------------------------------------
// Kernel 1: h0 = (1 + eps) * feature   (residual fused into accumulator init)
// ---------------------------------------------------------------------------
__global__ void gin_init(const float* __restrict__ feat,
                         const float* __restrict__ eps,
                         float* __restrict__ h0) {
    size_t i = (size_t)blockIdx.x * blockDim.x + threadIdx.x;   // float4 index
    const size_t n4 = (size_t)N_NODES * D / 4;
    if (i >= n4) return;
    const float s = 1.0f + eps[0];
    float4 v = ((const float4*)feat)[i];
    v.x *= s; v.y *= s; v.z *= s; v.w *= s;
    ((float4*)h0)[i] = v;
}

// ---------------------------------------------------------------------------
// Kernel 2: edge scatter-add. One wave per edge; lane covers dims
// {lane, lane+32, lane+64, lane+96} -> coalesced 128B bursts per wave.
// Both tables are L2-resident (25.6 MB each << 192 MB L2).
// ---------------------------------------------------------------------------
__global__ void gin_scatter(const float* __restrict__ feat,
                            const int* __restrict__ src,
                            const int* __restrict__ dst,
                            float* __restrict__ h0) {
    unsigned gid  = blockIdx.x * blockDim.x + threadIdx.x;
    unsigned e    = gid >> 5;
    unsigned lane = gid & 31u;
    if (e >= N_EDGES) return;
    const size_t soff = (size_t)src[e] * D;
    const size_t doff = (size_t)dst[e] * D;
#pragma unroll
    for (int j = 0; j < 4; ++j) {
        unsigned c = lane + 32u * j;
        atomicAdd(&h0[doff + c], feat[soff + c]);
    }
}

// ---------------------------------------------------------------------------
// Kernel 3: out = relu(h0 @ W^T + b) via V_WMMA_F32_16X16X4_F32.
// Block = 8 waves, owns a 32-wide N-group: W rows [g*32, g*32+32) staged in
// LDS (16 KB). Each wave owns one 16-row M-tile and TWO 16-col N-tiles:
//   per k-step: 1x global_load_b64 (A frag) + 2x ds_load_b64 (B frags)
//               + 2x v_wmma_f32_16x16x4_f32.
// Fragment layout per ISA: lane holds contiguous K pair at k0 + 2*(lane>=16).
// ---------------------------------------------------------------------------
__global__ void gin_gemm(const float* __restrict__ h0,
                         const float* __restrict__ W,
                         const float* __restrict__ bias,
                         float* __restrict__ out) {
    __shared__ float wlds[32 * D];           // 16 KB: 32 W rows for this N-group

    const int tid   = threadIdx.x;
    const int lane  = tid & 31;
    const int wave  = tid >> 5;              // 0..7
    const int half  = lane >> 4;             // 0: K pair {0,1}, 1: K pair {2,3}
    const int l16   = lane & 15;
    const int g     = blockIdx.y;            // N-group 0..3 (32 cols each)

    // Stage 32 W rows into LDS, coalesced float4.
    {
        const float4* wsrc = (const float4*)(W + (size_t)g * 32 * D);
        float4*       wdst = (float4*)wlds;
#pragma unroll
        for (int i = 0; i < 4; ++i)
            wdst[tid + 256 * i] = wsrc[tid + 256 * i];
    }
    __syncthreads();

    const int mtile = blockIdx.x * 8 + wave;
    if (mtile >= M_TILES) return;

    // Per-lane fragment base pointers (contiguous float2 at k0 + 2*half).
    const v2f* aptr  = (const v2f*)(h0 + (size_t)(mtile * 16 + l16) * D + 2 * half);
    const v2f* bptr0 = (const v2f*)(wlds + (l16)      * D + 2 * half);
    const v2f* bptr1 = (const v2f*)(wlds + (16 + l16) * D + 2 * half);

    v8f c0 = {0.f, 0.f, 0.f, 0.f, 0.f, 0.f, 0.f, 0.f};
    v8f c1 = {0.f, 0.f, 0.f, 0.f, 0.f, 0.f, 0.f, 0.f};

#pragma unroll
    for (int kk = 0; kk < 32; ++kk) {        // K = 128, step 4
        const v2f a  = aptr[kk * 2];         // global_load_b64
        const v2f b0 = bptr0[kk * 2];        // ds_load_b64
        const v2f b1 = bptr1[kk * 2];        // ds_load_b64
        c0 = __builtin_amdgcn_wmma_f32_16x16x4_f32(
                 false, a, false, b0, (short)0, c0, false, false);
        c1 = __builtin_amdgcn_wmma_f32_16x16x4_f32(
                 false, a, false, b1, (short)0, c1, false, false);
    }

    // Bias + ReLU + store. C layout: VGPR v -> row (v + 8*half), col l16.
    const int   ncol0 = g * 32 + l16;
    const float bv0   = bias[ncol0];
    const float bv1   = bias[ncol0 + 16];
    float* obase = out + (size_t)(mtile * 16 + half * 8) * D + ncol0;
#pragma unroll
    for (int v = 0; v < 8; ++v) {
        float r0 = c0[v] + bv0;
        float r1 = c1[v] + bv1;
        obase[(size_t)v * D]      = r0 > 0.f ? r0 : 0.f;
        obase[(size_t)v * D + 16] = r1 > 0.f ? r1 : 0.f;
    }
}

// ---------------------------------------------------------------------------
extern "C" void kernel_launch(void* const* d_in, const int* in_sizes, int n_in,
                              void* d_out, int out_size, void* d_ws, size_t ws_size,
                              hipStream_t stream) {
    const float* feat = (const float*)d_in[0];
    const int*   src  = (const int*)d_in[1];
    const int*   dst  = (const int*)d_in[2];
    const float* W    = (const float*)d_in[3];
    const float* bias = (const float*)d_in[4];
    const float* eps  = (const float*)d_in[5];
    float*       out  = (float*)d_out;
    float*       h0   = (float*)d_ws;        // N_NODES * D floats = 25.6 MB

    // 1) accumulator init with fused (1+eps) residual
    {
        const int n4 = N_NODES * D / 4;
        gin_init<<<(n4 + 255) / 256, 256, 0, stream>>>(feat, eps, h0);
    }
    // 2) edge scatter-add (one wave per edge)
    {
        const long long threads = (long long)N_EDGES * 32;
        gin_scatter<<<(unsigned)((threads + 255) / 256), 256, 0, stream>>>(
            feat, src, dst, h0);
    }
    // 3) WMMA f32 GEMM + bias + ReLU
    {
        dim3 grid((M_TILES + 7) / 8, D / 32);   // (391, 4)
        gin_gemm<<<grid, 256, 0, stream>>>(h0, W, bias, out);
    }
}